// MalwareDetectionRNN_32796370272874
// MI455X (gfx1250) — compile-verified
//
#include <hip/hip_runtime.h>

typedef float v2f __attribute__((ext_vector_type(2)));
typedef float v8f __attribute__((ext_vector_type(8)));

#define B_   32
#define T_   64
#define H_   200
#define NP   224    // N padded to 7 * 32 (GEMM) / 14 * 16 (scan)
#define SWST 202    // sW row stride (floats): gcd(202,64)=2 -> conflict-free b64
#define SHST 226    // sH row stride (floats): gcd(226,64)=2 -> conflict-free b64

static __device__ __forceinline__ v8f wmma_f32(v2f a, v2f b, v8f c) {
  // D = A(16x4,f32) * B(4x16,f32) + C(16x16,f32)
  return __builtin_amdgcn_wmma_f32_16x16x4_f32(false, a, false, b, (short)0, c,
                                               false, false);
}

// C[M x 200] = A[M x K] @ W[200 x K]^T + bias1[n] + bias2[n]
// One wave per 32x32 output macro-tile (2x2 register-blocked 16x16 WMMA tiles).
// grid = (M/32, 7), block = 32. Requires K % 8 == 0 (10000 and 200 both are).
__global__ __launch_bounds__(32) void gemm_bias_wmma(
    const float* __restrict__ A, const float* __restrict__ W,
    const float* __restrict__ bias1, const float* __restrict__ bias2,
    float* __restrict__ C, int K) {
  const int mtile = blockIdx.x * 32;
  const int ntile = blockIdx.y * 32;
  const int lane  = threadIdx.x;
  const int lid   = lane & 15;
  const int half  = lane >> 4;
  const int kb    = half * 2;              // hi half-wave carries K+2, K+3
  const int m0 = mtile + lid, m1 = mtile + 16 + lid;
  const int n0 = ntile + lid, n1 = ntile + 16 + lid;
  const bool nv0 = (n0 < H_), nv1 = (n1 < H_);
  const float* arow0 = A + (size_t)m0 * K;
  const float* arow1 = A + (size_t)m1 * K;
  const float* wrow0 = W + (size_t)(nv0 ? n0 : 0) * K;   // B[k][n] = W[n][k]
  const float* wrow1 = W + (size_t)(nv1 ? n1 : 0) * K;

  v8f acc00 = {}, acc01 = {}, acc10 = {}, acc11 = {};
  for (int k = 0; k < K; k += 8) {
    v2f a00 = *(const v2f*)(arow0 + k + kb);
    v2f a10 = *(const v2f*)(arow1 + k + kb);
    v2f b00 = *(const v2f*)(wrow0 + k + kb);
    v2f b10 = *(const v2f*)(wrow1 + k + kb);
    v2f a01 = *(const v2f*)(arow0 + k + 4 + kb);
    v2f a11 = *(const v2f*)(arow1 + k + 4 + kb);
    v2f b01 = *(const v2f*)(wrow0 + k + 4 + kb);
    v2f b11 = *(const v2f*)(wrow1 + k + 4 + kb);
    acc00 = wmma_f32(a00, b00, acc00);
    acc01 = wmma_f32(a00, b10, acc01);
    acc10 = wmma_f32(a10, b00, acc10);
    acc11 = wmma_f32(a10, b10, acc11);
    acc00 = wmma_f32(a01, b01, acc00);
    acc01 = wmma_f32(a01, b11, acc01);
    acc10 = wmma_f32(a11, b01, acc10);
    acc11 = wmma_f32(a11, b11, acc11);
  }

  const int r0 = mtile + (half << 3);        // VGPR v holds rows v (lo) / v+8 (hi)
  const int r1 = r0 + 16;
  if (nv0) {
    const float bias = bias1[n0] + bias2[n0];
    #pragma unroll
    for (int v = 0; v < 8; ++v) {
      C[(size_t)(r0 + v) * H_ + n0] = acc00[v] + bias;
      C[(size_t)(r1 + v) * H_ + n0] = acc10[v] + bias;
    }
  }
  if (nv1) {
    const float bias = bias1[n1] + bias2[n1];
    #pragma unroll
    for (int v = 0; v < 8; ++v) {
      C[(size_t)(r0 + v) * H_ + n1] = acc01[v] + bias;
      C[(size_t)(r1 + v) * H_ + n1] = acc11[v] + bias;
    }
  }
}

// Persistent-workgroup recurrent scan:
//   h_t = tanh(pre[:,t,:] + h_{t-1} @ Whh^T),  h_{-1} = 0
// W_hh (224 x 200, bank-conflict-free stride 202, pad rows zero) and h
// (32 x stride 226) live entirely in LDS (~205 KB of the WGP's 320 KB).
// 14 waves: each owns 1 M-tile x 2 N-tiles so the A(h) fragment is reused.
__global__ __launch_bounds__(448) void rnn_scan_wmma(
    const float* __restrict__ pre, const float* __restrict__ Whh,
    float* __restrict__ hseq) {
  extern __shared__ float smem[];
  float* sW = smem;                // [NP][SWST]
  float* sH = smem + NP * SWST;    // [B_][SHST]

  const int tid = threadIdx.x;
  for (int i = tid; i < NP * SWST; i += 448) {
    const int r = i / SWST, c = i - r * SWST;
    sW[i] = (r < H_ && c < H_) ? Whh[r * H_ + c] : 0.0f;
  }
  for (int i = tid; i < B_ * SHST; i += 448)
    sH[i] = 0.0f;
  __syncthreads();

  const int wave  = tid >> 5;               // 0..13
  const int lane  = tid & 31;
  const int lid   = lane & 15;
  const int half  = lane >> 4;
  const int kb    = half * 2;
  const int mtile = (wave & 1) * 16;
  const int nbase = (wave >> 1) * 32;       // 0,32,...,192
  const int m  = mtile + lid;
  const int n0 = nbase + lid;
  const int n1 = nbase + 16 + lid;
  const bool nv0 = (n0 < H_), nv1 = (n1 < H_);
  const float* sArow  = sH + m * SHST;
  const float* sBrow0 = sW + n0 * SWST;     // pad rows are zero
  const float* sBrow1 = sW + n1 * SWST;
  const int rbase = mtile + (half << 3);

  for (int t = 0; t < T_; ++t) {
    // hide the pre[:,t,:] fetch latency under the WMMA k-loop
    #pragma unroll
    for (int v = 0; v < 8; ++v)
      __builtin_prefetch(&pre[((size_t)(rbase + v) * T_ + t) * H_ + n0], 0, 1);

    v8f acc0 = {}, acc1 = {};
    for (int k = 0; k < H_; k += 8) {
      v2f a0 = *(const v2f*)(sArow + k + kb);
      v2f b00 = *(const v2f*)(sBrow0 + k + kb);
      v2f b10 = *(const v2f*)(sBrow1 + k + kb);
      v2f a1 = *(const v2f*)(sArow + k + 4 + kb);
      v2f b01 = *(const v2f*)(sBrow0 + k + 4 + kb);
      v2f b11 = *(const v2f*)(sBrow1 + k + 4 + kb);
      acc0 = wmma_f32(a0, b00, acc0);
      acc1 = wmma_f32(a0, b10, acc1);
      acc0 = wmma_f32(a1, b01, acc0);
      acc1 = wmma_f32(a1, b11, acc1);
    }

    float vals0[8], vals1[8];
    #pragma unroll
    for (int v = 0; v < 8; ++v) {
      const size_t prow = ((size_t)(rbase + v) * T_ + t) * H_;
      vals0[v] = tanhf(acc0[v] + (nv0 ? pre[prow + n0] : 0.0f));
      vals1[v] = tanhf(acc1[v] + (nv1 ? pre[prow + n1] : 0.0f));
    }
    __syncthreads();                        // all sH reads (DScnt) drained
    #pragma unroll
    for (int v = 0; v < 8; ++v) {
      const int row = rbase + v;
      const size_t hrow = ((size_t)row * T_ + t) * H_;
      if (nv0) { sH[row * SHST + n0] = vals0[v]; hseq[hrow + n0] = vals0[v]; }
      if (nv1) { sH[row * SHST + n1] = vals1[v]; hseq[hrow + n1] = vals1[v]; }
    }
    __syncthreads();
  }
}

// logits[b][c] = h_last[b] . fc_w[c] + fc_b[c]
__global__ __launch_bounds__(64) void fc_head(
    const float* __restrict__ hseq, const float* __restrict__ fcw,
    const float* __restrict__ fcb, float* __restrict__ out) {
  const int tid = threadIdx.x;
  if (tid >= B_ * 2) return;
  const int b = tid >> 1, c = tid & 1;
  const float* hrow = hseq + ((size_t)b * T_ + (T_ - 1)) * H_;
  const float* w = fcw + c * H_;
  float s = fcb[c];
  for (int k = 0; k < H_; ++k) s += hrow[k] * w[k];
  out[b * 2 + c] = s;
}

extern "C" void kernel_launch(void* const* d_in, const int* in_sizes, int n_in,
                              void* d_out, int out_size, void* d_ws, size_t ws_size,
                              hipStream_t stream) {
  const float* x    = (const float*)d_in[0];
  const float* Wih0 = (const float*)d_in[1];
  const float* Whh0 = (const float*)d_in[2];
  const float* bih0 = (const float*)d_in[3];
  const float* bhh0 = (const float*)d_in[4];
  const float* Wih1 = (const float*)d_in[5];
  const float* Whh1 = (const float*)d_in[6];
  const float* bih1 = (const float*)d_in[7];
  const float* bhh1 = (const float*)d_in[8];
  const float* Wih2 = (const float*)d_in[9];
  const float* Whh2 = (const float*)d_in[10];
  const float* bih2 = (const float*)d_in[11];
  const float* bhh2 = (const float*)d_in[12];
  const float* fcw  = (const float*)d_in[13];
  const float* fcb  = (const float*)d_in[14];

  float* buf0 = (float*)d_ws;                     // pre   (2048 x 200)
  float* buf1 = buf0 + (size_t)B_ * T_ * H_;      // h seq (2048 x 200)

  const dim3 ggrid((B_ * T_) / 32, NP / 32);      // (64, 7)
  const dim3 gblk(32);
  const size_t lds = (size_t)(NP * SWST + B_ * SHST) * sizeof(float);

  gemm_bias_wmma<<<ggrid, gblk, 0, stream>>>(x,    Wih0, bih0, bhh0, buf0, 10000);
  rnn_scan_wmma<<<1, 448, lds, stream>>>(buf0, Whh0, buf1);
  gemm_bias_wmma<<<ggrid, gblk, 0, stream>>>(buf1, Wih1, bih1, bhh1, buf0, H_);
  rnn_scan_wmma<<<1, 448, lds, stream>>>(buf0, Whh1, buf1);
  gemm_bias_wmma<<<ggrid, gblk, 0, stream>>>(buf1, Wih2, bih2, bhh2, buf0, H_);
  rnn_scan_wmma<<<1, 448, lds, stream>>>(buf0, Whh2, buf1);
  fc_head<<<1, 64, 0, stream>>>(buf1, fcw, fcb, (float*)d_out);
}